// SoftMesh_35313221107989
// MI455X (gfx1250) — compile-verified
//
#include <hip/hip_runtime.h>
#include <math.h>

// ---------------- problem constants ----------------
#define RESN    128
#define VOX     (128*128*128)          // 2^21 elements per volume
#define NBATCH  2
#define NCLS    4
#define NCOMP   3                      // classes kept (background dropped)
#define NVOL    (NBATCH*NCOMP)         // 6 volumes
#define NPTS_C  100000
#define LINES_PER_VOL 16384            // 128*128 lines per axis pass
#define TOTAL_LINES   (NVOL*LINES_PER_VOL)   // 98304
#define ROW_TILES     (TOTAL_LINES/16)       // 6144 (multiple of 4)
#define DFT_BLOCKS    (ROW_TILES/4)          // 1536 blocks of 4 waves
#define DFT_LDS_BYTES (2*16384*4)            // full cos+sin tables: 128 KB

typedef __attribute__((ext_vector_type(2))) float v2f;
typedef __attribute__((ext_vector_type(4))) float v4f;
typedef __attribute__((ext_vector_type(8))) float v8f;

// ---------------- WMMA helper (V_WMMA_F32_16X16X4_F32) ----------------
__device__ __forceinline__ v8f wmma_f32(v2f a, v2f b, v8f c) {
  // 8 args: (neg_a, A, neg_b, B, c_mod, C, reuse_a, reuse_b)
  return __builtin_amdgcn_wmma_f32_16x16x4_f32(false, a, false, b, (short)0, c,
                                               false, false);
}

// ---------------- utility ----------------
__global__ void zero_f32(float* __restrict__ p, size_t n) {
  size_t i = (size_t)blockIdx.x * blockDim.x + threadIdx.x;
  if (i < n) p[i] = 0.0f;
}

// DFT cos/sin tables (128x128 each) + derivative-kernel spectrum table S[128]
__global__ void init_tabs(float* __restrict__ Tc, float* __restrict__ Ts,
                          float* __restrict__ Sd) {
  int i = blockIdx.x * blockDim.x + threadIdx.x;
  const float TWOPI = 6.28318530717958647692f;
  if (i < 16384) {
    int w = i >> 7, n = i & 127;
    int m = (w * n) & 127;                  // angle reduced mod 2*pi
    float th = TWOPI * (float)m * (1.0f / 128.0f);
    Tc[i] = __cosf(th);
    Ts[i] = __sinf(th);
  } else if (i < 16384 + 128) {
    int w = i - 16384;
    // gaussian-derivative kernel: dk[t] = -t/sigma^2 * g(t)/sum(g), sigma=10, r=15
    float gs = 0.0f;
    for (int x = -15; x <= 15; ++x) gs += __expf(-0.5f * (float)(x * x) * 0.01f);
    float s = 0.0f;
    for (int t = 1; t <= 15; ++t) {
      float g  = __expf(-0.5f * (float)(t * t) * 0.01f) / gs;
      float kt = -((float)t * 0.01f) * g;   // dk at +t
      int m = (w * t) & 127;
      s += 2.0f * kt * __sinf(TWOPI * (float)m * (1.0f / 128.0f));
    }
    Sd[w] = s;    // cross-corr spectrum = i * Sd[w] (purely imaginary)
  }
}

// ---------------- trilinear corner stencil (shared by splat & sample) -------
__device__ __forceinline__ void tri_corners(float px, float py, float pz,
                                            int* __restrict__ flat,
                                            float* __restrict__ w) {
  float q[3];
  q[0] = (px + 1.0f) * 64.0f - 0.5f;   // ((p+1)*128 - 1)*0.5
  q[1] = (py + 1.0f) * 64.0f - 0.5f;
  q[2] = (pz + 1.0f) * 64.0f - 0.5f;
  int   q0[3];
  float fr[3];
#pragma unroll
  for (int c = 0; c < 3; ++c) {
    float f = floorf(q[c]);
    q0[c] = (int)f;
    fr[c] = q[c] - f;
  }
#pragma unroll
  for (int o = 0; o < 8; ++o) {
    int ox = (o >> 2) & 1, oy = (o >> 1) & 1, oz = o & 1;
    int ix = q0[0] + ox, iy = q0[1] + oy, iz = q0[2] + oz;
    float ww = (ox ? fr[0] : 1.0f - fr[0]) *
               (oy ? fr[1] : 1.0f - fr[1]) *
               (oz ? fr[2] : 1.0f - fr[2]);
    bool inb = (ix >= 0) & (ix < RESN) & (iy >= 0) & (iy < RESN) &
               (iz >= 0) & (iz < RESN);
    w[o]    = inb ? ww : 0.0f;
    flat[o] = inb ? ((ix * RESN + iy) * RESN + iz) : 0;
  }
}

// ---------------- softmax + trilinear splat ----------------
__global__ void splat_kernel(const float* __restrict__ logits,
                             const float* __restrict__ coords,
                             float* __restrict__ grid) {
  int j = blockIdx.x * blockDim.x + threadIdx.x;
  if (j >= NBATCH * NPTS_C) return;
  int b = j / NPTS_C, p = j - b * NPTS_C;

  float l[NCLS], mx = -1e30f;
#pragma unroll
  for (int c = 0; c < NCLS; ++c) {
    l[c] = logits[(size_t)(b * NCLS + c) * NPTS_C + p];
    mx = fmaxf(mx, l[c]);
  }
  float den = 0.0f;
#pragma unroll
  for (int c = 0; c < NCLS; ++c) { l[c] = __expf(l[c] - mx); den += l[c]; }
  float inv = 1.0f / den;
  float prob[NCOMP];
#pragma unroll
  for (int c = 0; c < NCOMP; ++c) prob[c] = l[c + 1] * inv;

  float px = coords[(size_t)(b * 3 + 0) * NPTS_C + p];
  float py = coords[(size_t)(b * 3 + 1) * NPTS_C + p];
  float pz = coords[(size_t)(b * 3 + 2) * NPTS_C + p];

  int   flat[8];
  float w[8];
  tri_corners(px, py, pz, flat, w);

#pragma unroll
  for (int o = 0; o < 8; ++o) {
    if (w[o] != 0.0f) {
#pragma unroll
      for (int c = 0; c < NCOMP; ++c) {
        atomicAdd(grid + ((size_t)(b * NCOMP + c) << 21) + flat[o],
                  prob[c] * w[o]);
      }
    }
  }
}

// ---------------- DFT pass geometry ----------------
template <int AXIS>
__device__ __forceinline__ size_t line_base(int batch, int l) {
  size_t b = (size_t)batch << 21;
  if (AXIS == 2) return b + ((size_t)l << 7);                 // contiguous z
  if (AXIS == 1) { int x = l >> 7, z = l & 127; return b + ((size_t)x << 14) + z; }
  return b + (size_t)l;                                       // axis 0 (x)
}

// ---------------- 128-point DFT along an axis, via V_WMMA_F32_16X16X4_F32 ---
// Each wave: 16 lines x all 128 output freqs (8 col-tiles of accumulators),
// so the A operand streams from HBM exactly once per pass.
// out_re = C*re + sgn*S*im ; out_im = C*im - sgn*S*re  (FWD: sgn=+1, else -1)
template <int AXIS, bool HAS_IM_IN, bool HAS_IM_OUT, bool FWD>
__global__ void __launch_bounds__(128) dft_pass(
    const float* __restrict__ in_re, const float* __restrict__ in_im,
    float* __restrict__ out_re, float* __restrict__ out_im,
    const float* __restrict__ Tc, const float* __restrict__ Ts) {
  constexpr int STRIDE = (AXIS == 2) ? 1 : ((AXIS == 1) ? RESN : RESN * RESN);

  extern __shared__ float smem[];      // 128 KB dynamic (320 KB/WGP on CDNA5)
  float* lds_c = smem;                 // [128 w][128 k] cos
  float* lds_s = smem + 16384;         // [128 w][128 k] sin

  // cooperative stage of the full transform tables (hits L2: tables are hot)
  for (int i = threadIdx.x * 4; i < 16384; i += 128 * 4) {
    *(v4f*)&lds_c[i] = *(const v4f*)&Tc[i];
    *(v4f*)&lds_s[i] = *(const v4f*)&Ts[i];
  }
  __syncthreads();

  const int wave = threadIdx.x >> 5;
  const int lane = threadIdx.x & 31;
  const int row_tile = blockIdx.x * 4 + wave;
  const int half = lane >> 4;
  const int lr   = lane & 15;

  const int line0   = row_tile * 16;             // never crosses a volume
  const int batch   = line0 >> 14;
  const int lA      = (line0 & 16383) + lr;
  const size_t baseA = line_base<AXIS>(batch, lA);

  const float* pAr = in_re + baseA + (size_t)(half * 2) * STRIDE;
  const float* pAi = HAS_IM_IN ? (in_im + baseA + (size_t)(half * 2) * STRIDE)
                               : nullptr;

  v8f acc_re[8];
  v8f acc_im[8];
#pragma unroll
  for (int ct = 0; ct < 8; ++ct) { acc_re[ct] = (v8f){}; acc_im[ct] = (v8f){}; }

#pragma unroll 2
  for (int k0 = 0; k0 < RESN; k0 += 4) {
    __builtin_prefetch(pAr + (size_t)16 * STRIDE, 0, 1);  // global_prefetch_b8

    v2f a_re, a_im;
    if (AXIS == 2) {
      a_re = *(const v2f*)pAr;                   // contiguous pair: b64 load
    } else {
      a_re.x = pAr[0];
      a_re.y = pAr[STRIDE];
    }
    if (HAS_IM_IN) {
      if (AXIS == 2) {
        a_im = *(const v2f*)pAi;
      } else {
        a_im.x = pAi[0];
        a_im.y = pAi[STRIDE];
      }
    }

    const int bidx = lr * RESN + k0 + half * 2;  // 8-byte aligned

#pragma unroll
    for (int ct = 0; ct < 8; ++ct) {             // ct*2048 floats -> ds imm off
      v2f b_c    = *(const v2f*)&lds_c[ct * 2048 + bidx];
      v2f b_sraw = *(const v2f*)&lds_s[ct * 2048 + bidx];
      v2f b_s  = FWD ? b_sraw : -b_sraw;         // sgn*S
      v2f b_sn = -b_s;

      acc_re[ct] = wmma_f32(a_re, b_c, acc_re[ct]);
      if (HAS_IM_IN) {
        acc_re[ct] = wmma_f32(a_im, b_s, acc_re[ct]);
        if (HAS_IM_OUT) {
          acc_im[ct] = wmma_f32(a_im, b_c, acc_im[ct]);
          acc_im[ct] = wmma_f32(a_re, b_sn, acc_im[ct]);
        }
      } else if (HAS_IM_OUT) {                   // real input: im' = -sgn*S*re
        acc_im[ct] = wmma_f32(a_re, b_sn, acc_im[ct]);
      }
    }

    pAr += (size_t)4 * STRIDE;
    if (HAS_IM_IN) pAi += (size_t)4 * STRIDE;
  }

  // epilogue: one base address per output row, col-tiles via immediate offsets
#pragma unroll
  for (int v = 0; v < 8; ++v) {                  // D layout: row = v + 8*half
    int lrow = (line0 & 16383) + v + half * 8;
    size_t ob0 = line_base<AXIS>(batch, lrow) + (size_t)lr * STRIDE;
    float* por = out_re + ob0;
    float* poi = HAS_IM_OUT ? (out_im + ob0) : nullptr;
#pragma unroll
    for (int ct = 0; ct < 8; ++ct) {             // +ct*16*STRIDE (imm offset)
      por[(size_t)(ct * 16) * STRIDE] = acc_re[ct][v];
      if (HAS_IM_OUT) poi[(size_t)(ct * 16) * STRIDE] = acc_im[ct][v];
    }
  }
}

// ---------------- fused spectral multiplier (smooth+div+Poisson+1/N^3) -----
__global__ void specmul(float* __restrict__ Qre, float* __restrict__ Qim,
                        const float* __restrict__ Sd) {
  size_t v = (size_t)blockIdx.x * blockDim.x + threadIdx.x;
  if (v >= (size_t)VOX) return;
  int i = (int)(v >> 14), j = (int)((v >> 7) & 127), k = (int)(v & 127);
  float fx = (i < 64) ? (float)i : (float)(i - 128);
  float fy = (j < 64) ? (float)j : (float)(j - 128);
  float fz = (k < 64) ? (float)k : (float)(k - 128);
  float f2 = fx * fx + fy * fy + fz * fz;
  float M = 0.0f;
  if (v != 0) {
    const float TWOPI = 6.28318530717958647692f;
    float G   = __expf(-0.5f * 0.0244140625f * f2);   // (2*sigma/128)^2
    float lap = -(TWOPI * TWOPI) * f2;
    float num = TWOPI * (fx * Sd[i] + fy * Sd[j] + fz * Sd[k]);
    M = (num * G / lap) * (1.0f / (float)VOX);        // fold inverse-FFT scale
  }
#pragma unroll
  for (int bc = 0; bc < NVOL; ++bc) {
    size_t o = ((size_t)bc << 21) + v;
    Qre[o] *= M;
    Qim[o] *= M;
  }
}

// ---------------- trilinear sample of phi at input points -> mean ----------
__global__ void sample_mean(const float* __restrict__ coords,
                            const float* __restrict__ phi,
                            float* __restrict__ fvsum) {
  int j = blockIdx.x * blockDim.x + threadIdx.x;
  if (j >= NBATCH * NPTS_C) return;
  int b = j / NPTS_C, p = j - b * NPTS_C;
  float px = coords[(size_t)(b * 3 + 0) * NPTS_C + p];
  float py = coords[(size_t)(b * 3 + 1) * NPTS_C + p];
  float pz = coords[(size_t)(b * 3 + 2) * NPTS_C + p];
  int   flat[8];
  float w[8];
  tri_corners(px, py, pz, flat, w);
#pragma unroll
  for (int c = 0; c < NCOMP; ++c) {
    const float* vol = phi + ((size_t)(b * NCOMP + c) << 21);
    float s = 0.0f;
#pragma unroll
    for (int o = 0; o < 8; ++o) s += w[o] * vol[flat[o]];
    atomicAdd(&fvsum[b * NCOMP + c], s);
  }
}

// ---------------- shift by mean-sample, normalize by |phi(0,0,0)| ----------
__global__ void finalize(const float* __restrict__ phi,
                         const float* __restrict__ fvsum,
                         float* __restrict__ out) {
  size_t i = (size_t)blockIdx.x * blockDim.x + threadIdx.x;
  if (i >= (size_t)NVOL * VOX) return;
  int bc = (int)(i >> 21);
  float fv  = fvsum[bc] * (1.0f / (float)NPTS_C);
  float p00 = phi[(size_t)bc << 21] - fv;
  float sc  = -0.5f / fabsf(p00);
  out[i] = (phi[i] - fv) * sc;
}

// ---------------- host orchestration ----------------
extern "C" void kernel_launch(void* const* d_in, const int* in_sizes, int n_in,
                              void* d_out, int out_size, void* d_ws,
                              size_t ws_size, hipStream_t stream) {
  const float* logits = (const float*)d_in[0];   // (2,4,100000) f32
  const float* coords = (const float*)d_in[1];   // (2,3,100000) f32
  float* out = (float*)d_out;                    // (6,128,128,128) f32

  float* ws = (float*)d_ws;
  const size_t NBf = (size_t)NVOL * VOX;         // 12,582,912 floats / buffer
  float* Pre = ws;
  float* Pim = Pre + NBf;
  float* Qre = Pim + NBf;
  float* Qim = Qre + NBf;
  float* Tc  = Qim + NBf;                        // 128x128 cos
  float* Ts  = Tc + 16384;                       // 128x128 sin
  float* Sd  = Ts + 16384;                       // deriv spectrum, 128
  float* fvs = Sd + 128;                         // 6 accumulators

  // init
  zero_f32<<<(unsigned)((NBf + 255) / 256), 256, 0, stream>>>(Pre, NBf);
  zero_f32<<<1, 32, 0, stream>>>(fvs, NVOL);
  init_tabs<<<(16384 + 128 + 255) / 256, 256, 0, stream>>>(Tc, Ts, Sd);

  // softmax + splat into Pre (6 real volumes)
  splat_kernel<<<(NBATCH * NPTS_C + 255) / 256, 256, 0, stream>>>(logits,
                                                                  coords, Pre);

  // forward 3D DFT (real -> complex), ping-pong P <-> Q
  dft_pass<2, false, true, true>
      <<<DFT_BLOCKS, 128, DFT_LDS_BYTES, stream>>>(Pre, nullptr, Qre, Qim,
                                                   Tc, Ts);
  dft_pass<1, true, true, true>
      <<<DFT_BLOCKS, 128, DFT_LDS_BYTES, stream>>>(Qre, Qim, Pre, Pim, Tc, Ts);
  dft_pass<0, true, true, true>
      <<<DFT_BLOCKS, 128, DFT_LDS_BYTES, stream>>>(Pre, Pim, Qre, Qim, Tc, Ts);

  // fused spectral operator (G-smooth + derivative + divergence + Poisson)
  specmul<<<VOX / 256, 256, 0, stream>>>(Qre, Qim, Sd);

  // inverse 3D DFT (complex -> real phi in Pre)
  dft_pass<2, true, true, false>
      <<<DFT_BLOCKS, 128, DFT_LDS_BYTES, stream>>>(Qre, Qim, Pre, Pim, Tc, Ts);
  dft_pass<1, true, true, false>
      <<<DFT_BLOCKS, 128, DFT_LDS_BYTES, stream>>>(Pre, Pim, Qre, Qim, Tc, Ts);
  dft_pass<0, true, false, false>
      <<<DFT_BLOCKS, 128, DFT_LDS_BYTES, stream>>>(Qre, Qim, Pre, nullptr,
                                                   Tc, Ts);

  // shift + scale
  sample_mean<<<(NBATCH * NPTS_C + 255) / 256, 256, 0, stream>>>(coords, Pre,
                                                                 fvs);
  finalize<<<(unsigned)((NBf + 255) / 256), 256, 0, stream>>>(Pre, fvs, out);
}